// PMHCProjectionHead_4080218931904
// MI455X (gfx1250) — compile-verified
//
#include <hip/hip_runtime.h>

typedef __attribute__((ext_vector_type(2))) float v2f;
typedef __attribute__((ext_vector_type(8))) float v8f;

#define WMMA_F32(a, b, c) \
  __builtin_amdgcn_wmma_f32_16x16x4_f32(false, (a), false, (b), (short)0, (c), false, false)

// ---- CDNA5 async memory->LDS path (probe-confirmed prototype:
//      (AS1 int* gaddr, AS3 int* ldsaddr, imm offset, imm cpol)) -------------
#if __has_builtin(__builtin_amdgcn_global_load_async_to_lds_b32)
#define HAVE_ASYNC_LDS 1
typedef __attribute__((address_space(1))) int gbl_i32;
typedef __attribute__((address_space(3))) int lds_i32;
#else
#define HAVE_ASYNC_LDS 0
#endif

__device__ __forceinline__ void async_copy_f32(const float* g, float* l) {
#if HAVE_ASYNC_LDS
  __builtin_amdgcn_global_load_async_to_lds_b32((gbl_i32*)g, (lds_i32*)l, 0, 0);
#else
  (void)g; (void)l;
#endif
}

__device__ __forceinline__ void wait_async_cnt0() {
#if __has_builtin(__builtin_amdgcn_s_wait_asynccnt)
  __builtin_amdgcn_s_wait_asynccnt(0);
#else
  asm volatile("s_wait_asynccnt 0x0" ::: "memory");
#endif
}

namespace {
constexpr int kD  = 960;   // embedding dim
constexpr int kRL = 16;    // low-rank L
constexpr int kRD = 32;    // low-rank D
constexpr int kCR = 32;    // l-chunk rows
constexpr int kDC = 64;    // d-chunk cols
constexpr int kND = kD / kDC;  // 15 d-chunks
constexpr int kB  = 512;   // batch
}

// ---------------------------------------------------------------------------
// Encoder core: per batch sample b compute
//   V[p, d] = sum_l (A_c[l,p] * mask[b,l]) * emb[b,l,d]     (streamed once)
//   U[p, r] = sum_d V[p, d] * B_c[d, r]
// stored as U[b, p*32 + r] (matches reference reshape order).
// Block = 128 threads (4 waves). One sample per workgroup. L is compile-time.
// All flat offsets fit in int32 (max 512*384*960 < 2^31).
// ---------------------------------------------------------------------------
template <int L>
__global__ __launch_bounds__(128)
void encode_kernel(const float* __restrict__ emb, const int* __restrict__ mask,
                   const float* __restrict__ Ac,  const float* __restrict__ Bc,
                   float* __restrict__ U)
{
  constexpr int  nLC   = (L + kCR - 1) / kCR;
  constexpr int  Lpad  = nLC * kCR;
  constexpr bool kFull = (L % kCR) == 0;   // every l-chunk fully in-range

  __shared__ float sXT[kDC][kCR + 2];     // X chunk transposed [d][l], stride 34
  __shared__ float sAmT[kRL][Lpad + 2];   // masked A^T [p][l], whole sample
  __shared__ float sV[kRL][kDC + 2];      // V d-slice [p][d], stride 66
  __shared__ float sBT[kRD][kDC + 2];     // B_c^T slice [r][d], stride 66

  const int b     = blockIdx.x;
  const int tid   = threadIdx.x;
  const int wave  = tid >> 5;
  const int lane  = tid & 31;
  const int lhalf = lane >> 4;   // selects K pair within fragment
  const int l16   = lane & 15;

  const int embBase = b * (L * kD);

  // Build masked A^T once per sample (branch-free clamp+select)
  for (int idx = tid; idx < kRL * Lpad; idx += 128) {
    int p = idx & 15, l = idx >> 4;
    int lcl = (l < L) ? l : (L - 1);
    float v = Ac[lcl * kRL + p] * (float)mask[b * L + lcl];
    sAmT[p][l] = (l < L) ? v : 0.f;
  }

  v8f accU = {};  // U tile (waves 0,1 only)

  for (int dc = 0; dc < kND; ++dc) {
    const int d0 = dc * kDC;
    v8f accV = {};  // this wave's 16x16 V tile for d in [d0+wave*16, +16)

    for (int lc = 0; lc < nLC; ++lc) {
      const int l0 = lc * kCR;
      __syncthreads();  // previous readers of sXT done (also fences sAmT build)

      if constexpr (kFull && HAVE_ASYNC_LDS) {
        // transpose-on-the-fly: per-lane scattered LDS dest, no VGPR round-trip
        #pragma unroll
        for (int i = 0; i < 16; ++i) {
          int idx = tid + 128 * i;      // 0..2047 = 32l x 64d (coalesced in d)
          int l = idx >> 6, dd = idx & 63;
          const float* gp = emb + (embBase + (l0 + l) * kD + d0 + dd);
          async_copy_f32(gp, &sXT[dd][l]);
          if (((i & 7) == 0) && (d0 + kDC < kD))
            __builtin_prefetch(gp + kDC, 0, 1);   // global_prefetch_b8
        }
        wait_async_cnt0();
      } else {
        #pragma unroll
        for (int i = 0; i < 16; ++i) {
          int idx = tid + 128 * i;
          int l = idx >> 6, dd = idx & 63;
          int lg  = l0 + l;
          int lcl = (lg < L) ? lg : (L - 1);
          float v = emb[embBase + lcl * kD + d0 + dd];
          sXT[dd][l] = (lg < L) ? v : 0.f;
        }
      }
      __syncthreads();

      // stage-1 WMMA: V(16x16) += A_m^T(16 x 32) @ X(32 x 16), K = l
      {
        const float* ap = &sAmT[l16][l0 + 2 * lhalf];
        const float* bp = &sXT[wave * 16 + l16][2 * lhalf];
        #pragma unroll
        for (int kk = 0; kk < kCR; kk += 4) {
          v2f a  = *(const v2f*)(ap + kk);
          v2f bb = *(const v2f*)(bp + kk);
          accV = WMMA_F32(a, bb, accV);
        }
      }
    }

    __syncthreads();  // stage-2 readers of sV/sBT from previous dc done

    // dump V tiles: D-layout lane holds (m = 8*lhalf + j, n = l16)
    #pragma unroll
    for (int j = 0; j < 8; ++j)
      sV[8 * lhalf + j][wave * 16 + l16] = accV[j];

    // stage B_c^T slice: sBT[r][d] = Bc[d0+d, r]  (coalesced in r)
    #pragma unroll
    for (int i = 0; i < 16; ++i) {
      int idx = tid + 128 * i;          // 0..2047 = 64d x 32r
      int d = idx >> 5, r = idx & 31;
      sBT[r][d] = Bc[(d0 + d) * kRD + r];
    }
    __syncthreads();

    // stage-2 WMMA: U(16x32) += V(16 x 64) @ Bc(64 x 32), K = d
    if (wave < 2) {
      const float* ap = &sV[l16][2 * lhalf];
      const float* bp = &sBT[wave * 16 + l16][2 * lhalf];
      #pragma unroll
      for (int kk = 0; kk < kDC; kk += 4) {
        v2f a  = *(const v2f*)(ap + kk);
        v2f bb = *(const v2f*)(bp + kk);
        accU = WMMA_F32(a, bb, accU);
      }
    }
  }

  if (wave < 2) {
    #pragma unroll
    for (int j = 0; j < 8; ++j) {
      int p = 8 * lhalf + j;
      int r = wave * 16 + l16;
      U[b * (kRL * kRD) + p * kRD + r] = accU[j];
    }
  }
}

// ---------------------------------------------------------------------------
// Generic WMMA GEMM with epilogue: C = A(MxK) @ W(KxN)
//   mode 0: plain store
//   mode 1: v = (v + bias) * (g * rsqrt(1+eps)) + beta; relu
//   mode 2: v = v + bias
// Stores to Cout[m*ldc + col_off + n]. Block 128 (2x2 waves -> 32x32 tile).
// Branch-free clamp+select staging; pure int32 indexing.
// ---------------------------------------------------------------------------
__global__ __launch_bounds__(128)
void gemm_kernel(const float* __restrict__ A, const float* __restrict__ W,
                 const float* __restrict__ bias, const float* __restrict__ g,
                 const float* __restrict__ beta, float* __restrict__ Cout,
                 int N, int K, int ldc, int col_off, int mode)
{
  __shared__ float sA[32][34];    // [m][k]
  __shared__ float sWT[32][34];   // [n][k]

  const int tid   = threadIdx.x;
  const int wave  = tid >> 5;
  const int lane  = tid & 31;
  const int lhalf = lane >> 4;
  const int l16   = lane & 15;
  const int row0  = blockIdx.y * 32;
  const int col0  = blockIdx.x * 32;
  const int m0    = (wave >> 1) * 16;
  const int n0    = (wave & 1) * 16;

  v8f acc = {};

  for (int k0 = 0; k0 < K; k0 += 32) {
    __syncthreads();
    #pragma unroll
    for (int i = 0; i < 8; ++i) {
      int idx = tid + 128 * i;          // 0..1023
      int m  = idx >> 5, k = idx & 31;  // A tile: coalesced in k
      int kA = k0 + k;
      int kAc = (kA < K) ? kA : (K - 1);
      float va = A[(row0 + m) * K + kAc];
      sA[m][k] = (kA < K) ? va : 0.f;

      int kw = idx >> 5, n = idx & 31;  // W tile: coalesced in n
      int kW = k0 + kw;
      int kWc = (kW < K) ? kW : (K - 1);
      int nG  = col0 + n;
      int nGc = (nG < N) ? nG : (N - 1);
      float vw = W[kWc * N + nGc];
      sWT[n][kw] = (kW < K && nG < N) ? vw : 0.f;
    }
    __syncthreads();

    const float* ap = &sA[m0 + l16][2 * lhalf];
    const float* bp = &sWT[n0 + l16][2 * lhalf];
    #pragma unroll
    for (int kk = 0; kk < 32; kk += 4) {
      v2f a = *(const v2f*)(ap + kk);
      v2f b = *(const v2f*)(bp + kk);
      acc = WMMA_F32(a, b, acc);
    }
  }

  constexpr float kBNScale = 0.9999950000374997f;  // 1/sqrt(1 + 1e-5)
  const int n = col0 + n0 + l16;
  if (n < N) {
    #pragma unroll
    for (int j = 0; j < 8; ++j) {
      int m = row0 + m0 + 8 * lhalf + j;
      float v = acc[j];
      if (mode >= 1) v += bias[n];
      if (mode == 1) {
        v = v * (g[n] * kBNScale) + beta[n];
        v = v > 0.f ? v : 0.f;
      }
      Cout[m * ldc + col_off + n] = v;
    }
  }
}

// ---------------------------------------------------------------------------
extern "C" void kernel_launch(void* const* d_in, const int* in_sizes, int n_in,
                              void* d_out, int out_size, void* d_ws, size_t ws_size,
                              hipStream_t stream) {
  (void)in_sizes; (void)n_in; (void)out_size; (void)ws_size;

  const float* emb_P  = (const float*)d_in[0];
  const float* emb_H  = (const float*)d_in[1];
  const int*   mask_P = (const int*)  d_in[2];
  const int*   mask_H = (const int*)  d_in[3];
  const float* B_cP   = (const float*)d_in[4];
  const float* A_cP   = (const float*)d_in[5];
  const float* H_cP   = (const float*)d_in[6];
  const float* W1P    = (const float*)d_in[7];
  const float* b1P    = (const float*)d_in[8];
  const float* gP     = (const float*)d_in[9];
  const float* betaP  = (const float*)d_in[10];
  const float* W2P    = (const float*)d_in[11];
  const float* b2P    = (const float*)d_in[12];
  const float* B_cH   = (const float*)d_in[13];
  const float* A_cH   = (const float*)d_in[14];
  const float* H_cH   = (const float*)d_in[15];
  const float* W1H    = (const float*)d_in[16];
  const float* b1H    = (const float*)d_in[17];
  const float* gH     = (const float*)d_in[18];
  const float* betaH  = (const float*)d_in[19];
  const float* W2H    = (const float*)d_in[20];
  const float* b2H    = (const float*)d_in[21];

  float* out = (float*)d_out;
  float* ws  = (float*)d_ws;

  const int dP = 358, dH = 154, dMod = 512;
  float* UP = ws;                       // 512*512
  float* UH = UP + kB * dMod;           // 512*512
  float* zP = UH + kB * dMod;           // 512*358
  float* hP = zP + kB * dP;             // 512*358
  float* zH = hP + kB * dP;             // 512*154
  float* hH = zH + kB * dH;             // 512*154  (total 4 MiB fp32)

  dim3 blk(128);

  // encoders (HBM-bound stream of emb_H)
  encode_kernel<16><<<kB, blk, 0, stream>>>(emb_P, mask_P, A_cP, B_cP, UP);
  encode_kernel<384><<<kB, blk, 0, stream>>>(emb_H, mask_H, A_cH, B_cH, UH);

  // P head: z = U@H_c ; h = relu(BN(z@W1+b1)) ; out[:, 0:358] = h@W2+b2
  gemm_kernel<<<dim3((dP + 31) / 32, kB / 32), blk, 0, stream>>>(
      UP, H_cP, nullptr, nullptr, nullptr, zP, dP, dMod, dP, 0, 0);
  gemm_kernel<<<dim3((dP + 31) / 32, kB / 32), blk, 0, stream>>>(
      zP, W1P, b1P, gP, betaP, hP, dP, dP, dP, 0, 1);
  gemm_kernel<<<dim3((dP + 31) / 32, kB / 32), blk, 0, stream>>>(
      hP, W2P, b2P, nullptr, nullptr, out, dP, dP, 512, 0, 2);

  // H head: out[:, 358:512]
  gemm_kernel<<<dim3((dH + 31) / 32, kB / 32), blk, 0, stream>>>(
      UH, H_cH, nullptr, nullptr, nullptr, zH, dH, dMod, dH, 0, 0);
  gemm_kernel<<<dim3((dH + 31) / 32, kB / 32), blk, 0, stream>>>(
      zH, W1H, b1H, gH, betaH, hH, dH, dH, dH, 0, 1);
  gemm_kernel<<<dim3((dH + 31) / 32, kB / 32), blk, 0, stream>>>(
      hH, W2H, b2H, nullptr, nullptr, out, dH, dH, 512, 358, 2);
}